// Attention_6047313953243
// MI455X (gfx1250) — compile-verified
//
#include <hip/hip_runtime.h>
#include <hip/hip_bf16.h>
#include <math.h>

typedef __attribute__((ext_vector_type(2))) float v2f;
typedef __attribute__((ext_vector_type(8))) float v8f;

#define BATCH 4
#define CIN   128
#define OUTC  64
#define HIN   512
#define WIN   512
#define HP    64
#define WP    64
#define SPP   (HP*WP)   // 4096 pooled pixels per batch image

// ---------------------------------------------------------------------------
// Kernel 1: 8x8 max pool (three stacked 2x2 pools == one 8x8 pool).
// Output layout p[b][s][c] (channel-contiguous) so WMMA A-fragments are
// contiguous float2 loads along K.
// ---------------------------------------------------------------------------
__global__ __launch_bounds__(256)
void pool8_kernel(const float* __restrict__ x, float* __restrict__ p) {
    int idx = blockIdx.x * blockDim.x + threadIdx.x;   // 4*128*4096 = 2^21
    int sx = idx & 63;
    int sy = (idx >> 6) & 63;
    int c  = (idx >> 12) & 127;
    int b  = idx >> 19;
    const float* base = x + ((size_t)(b * CIN + c) * HIN + sy * 8) * WIN + sx * 8;
    float m = -INFINITY;
#pragma unroll
    for (int dy = 0; dy < 8; ++dy) {
        float4 r0 = *(const float4*)(base + (size_t)dy * WIN);
        float4 r1 = *(const float4*)(base + (size_t)dy * WIN + 4);
        m = fmaxf(m, fmaxf(fmaxf(r0.x, r0.y), fmaxf(r0.z, r0.w)));
        m = fmaxf(m, fmaxf(fmaxf(r1.x, r1.y), fmaxf(r1.z, r1.w)));
    }
    p[((size_t)b * SPP + sy * WP + sx) * CIN + c] = m;
}

// ---------------------------------------------------------------------------
// Kernel 2: dual 1x1 conv as WMMA GEMM.  D = A(16x4 f32) * B(4x16 f32) + C,
// accumulated over K=128 in 32 steps.  Each wave: one 16-row spatial tile x
// all 64 output channels for BOTH Wk and Wv (8 v8f accumulators).
// A layout (ISA 7.12.2, 32-bit A 16x4): lane<16 -> M=lane, K={k0,k0+1};
// lane>=16 -> M=lane-16, K={k0+2,k0+3}  => one contiguous v2f load.
// B (4x16 K-major) mirrors the split: VGPR0={K0,K2}, VGPR1={K1,K3}.
// ---------------------------------------------------------------------------
__global__ __launch_bounds__(256)
void gemm_wmma_kernel(const float* __restrict__ p,
                      const float* __restrict__ Wk, const float* __restrict__ bk,
                      const float* __restrict__ Wv, const float* __restrict__ bv,
                      float* __restrict__ kq, float* __restrict__ vq) {
    __shared__ float sW[2 * OUTC * CIN];               // 64 KB: Wk then Wv
    int tid = threadIdx.x;

    const float4* wk4 = (const float4*)Wk;
    const float4* wv4 = (const float4*)Wv;
    float4* s4 = (float4*)sW;
#pragma unroll
    for (int i = 0; i < (OUTC * CIN / 4) / 256; ++i) { // 8 iters
        s4[tid + i * 256]        = wk4[tid + i * 256];
        s4[2048 + tid + i * 256] = wv4[tid + i * 256];
    }
    __syncthreads();

    int lane  = tid & 31;
    int wave  = tid >> 5;
    int mtile = blockIdx.x * 8 + wave;                 // 0..1023
    int b     = mtile >> 8;                            // 256 M-tiles per batch
    int m0    = (mtile & 255) * 16;                    // spatial base
    int lrow  = lane & 15;
    int lk    = (lane >> 4) * 2;                       // 0 or 2

    const float* abase = p + ((size_t)b * SPP + m0 + lrow) * CIN + lk;

    v8f acck[4] = {};
    v8f accv[4] = {};

#pragma unroll 4
    for (int k0 = 0; k0 < CIN; k0 += 4) {
        v2f a = *(const v2f*)(abase + k0);
#pragma unroll
        for (int t = 0; t < 4; ++t) {
            int n = t * 16 + lrow;
            v2f bkf = *(const v2f*)&sW[n * CIN + k0 + lk];
            v2f bvf = *(const v2f*)&sW[OUTC * CIN + n * CIN + k0 + lk];
            acck[t] = __builtin_amdgcn_wmma_f32_16x16x4_f32(
                false, a, false, bkf, (short)0, acck[t], false, false);
            accv[t] = __builtin_amdgcn_wmma_f32_16x16x4_f32(
                false, a, false, bvf, (short)0, accv[t], false, false);
        }
    }

    // Epilogue: D layout (32-bit C/D 16x16): VGPR r, lane<16 -> M=r, N=lane;
    // lane>=16 -> M=r+8, N=lane-16.  Store [b][s][o]: 16 lanes contiguous.
#pragma unroll
    for (int t = 0; t < 4; ++t) {
        int n = t * 16 + lrow;
        float kb = bk[n];
        float vb = bv[n];
#pragma unroll
        for (int r = 0; r < 8; ++r) {
            int s = m0 + r + 8 * (lane >> 4);
            size_t oidx = ((size_t)b * SPP + s) * OUTC + n;
            kq[oidx] = acck[t][r] + kb;
            vq[oidx] = accv[t][r] + vb;
        }
    }
}

// ---------------------------------------------------------------------------
// Kernel 3: vmean[b,o] = mean over spatial of v.  One block per (b,o).
// ---------------------------------------------------------------------------
__global__ __launch_bounds__(256)
void vmean_kernel(const float* __restrict__ vq, float* __restrict__ vmean) {
    __shared__ float red[256];
    int bo = blockIdx.x;            // b*64 + o
    int b = bo >> 6, o = bo & 63;
    float s = 0.0f;
    for (int i = threadIdx.x; i < SPP; i += 256)
        s += vq[((size_t)b * SPP + i) * OUTC + o];
    red[threadIdx.x] = s;
    __syncthreads();
    for (int w = 128; w > 0; w >>= 1) {
        if (threadIdx.x < w) red[threadIdx.x] += red[threadIdx.x + w];
        __syncthreads();
    }
    if (threadIdx.x == 0) vmean[bo] = red[0] * (1.0f / (float)SPP);
}

// ---------------------------------------------------------------------------
// Kernel 4: out = bilinear_upsample_x8(vmean * k), half-pixel, edge-clamped
// (== F.interpolate(align_corners=False) == jax.image.resize bilinear).
// One thread produces 4 consecutive x outputs -> float4 store (256 MB stream).
// ---------------------------------------------------------------------------
__global__ __launch_bounds__(256)
void upsample_kernel(const float* __restrict__ kq, const float* __restrict__ vmean,
                     float* __restrict__ out) {
    int gid = blockIdx.x * blockDim.x + threadIdx.x;   // 2^24 threads
    int xg = (gid & 127) * 4;
    int y  = (gid >> 7) & 511;
    int o  = (gid >> 16) & 63;
    int b  = gid >> 22;

    float vm = vmean[b * OUTC + o];

    float sy = (y + 0.5f) * 0.125f - 0.5f;
    int   fy = (int)floorf(sy);
    float wy = sy - (float)fy;
    int iy0 = min(HP - 1, max(0, fy));
    int iy1 = min(HP - 1, max(0, fy + 1));

    const float* kb = kq + (size_t)b * SPP * OUTC + o;
    float r[4];
#pragma unroll
    for (int j = 0; j < 4; ++j) {
        int   xx = xg + j;
        float sx = (xx + 0.5f) * 0.125f - 0.5f;
        int   fx = (int)floorf(sx);
        float wx = sx - (float)fx;
        int ix0 = min(WP - 1, max(0, fx));
        int ix1 = min(WP - 1, max(0, fx + 1));
        float a00 = kb[(size_t)(iy0 * WP + ix0) * OUTC];
        float a01 = kb[(size_t)(iy0 * WP + ix1) * OUTC];
        float a10 = kb[(size_t)(iy1 * WP + ix0) * OUTC];
        float a11 = kb[(size_t)(iy1 * WP + ix1) * OUTC];
        float top = a00 + wx * (a01 - a00);
        float bot = a10 + wx * (a11 - a10);
        r[j] = vm * (top + wy * (bot - top));
    }
    size_t oidx = (((size_t)(b * OUTC + o) * 512) + y) * 512 + xg;
    *(float4*)(out + oidx) = make_float4(r[0], r[1], r[2], r[3]);
}

// ---------------------------------------------------------------------------
extern "C" void kernel_launch(void* const* d_in, const int* in_sizes, int n_in,
                              void* d_out, int out_size, void* d_ws, size_t ws_size,
                              hipStream_t stream) {
    const float* x  = (const float*)d_in[0];
    const float* Wk = (const float*)d_in[1];
    const float* bk = (const float*)d_in[2];
    const float* Wv = (const float*)d_in[3];
    const float* bv = (const float*)d_in[4];
    float* out = (float*)d_out;

    // Workspace layout (floats): p[4*4096*128] | kq[4*4096*64] | vq[...] | vmean[256]
    float* p   = (float*)d_ws;
    float* kq  = p  + (size_t)BATCH * SPP * CIN;
    float* vq  = kq + (size_t)BATCH * SPP * OUTC;
    float* vme = vq + (size_t)BATCH * SPP * OUTC;

    pool8_kernel   <<<(BATCH * CIN * SPP) / 256, 256, 0, stream>>>(x, p);
    gemm_wmma_kernel<<<128, 256, 0, stream>>>(p, Wk, bk, Wv, bv, kq, vq);
    vmean_kernel   <<<BATCH * OUTC, 256, 0, stream>>>(vq, vme);
    upsample_kernel<<<(BATCH * OUTC * 512 * 128) / 256, 256, 0, stream>>>(kq, vme, out);
}